// SinkhornSolver_8091718385859
// MI455X (gfx1250) — compile-verified
//
#include <hip/hip_runtime.h>

// ---------------------------------------------------------------------------
// Sinkhorn (entropic OT) on MI455X / gfx1250.
//
// Scaling-form reformulation:
//   E  = exp(-C / eps)                          (one pass, stored in d_out)
//   wv = 1
//   repeat 100x:  wu = a ./ (E  @ wv)           (row pass, WMMA reduction)
//                 wv = b ./ (E' @ wu)           (col pass, WMMA reduction)
//   pi = diag(wu) * E * diag(wv)                (in-place on d_out)
//
// E (128 MB) fits in the 192 MB L2 -> the 200 streaming passes are L2-bound.
// Reductions run on the matrix pipe via V_WMMA_F32_16X16X4_F32. Row-pass
// tiles stream global->LDS with global_load_async_to_lds_b128, DOUBLE
// BUFFERED: prefetch chunk N+1, wait ASYNCcnt<=4 (async completions retire
// in order, so the current buffer's 4 transfers are done), compute chunk N.
// ---------------------------------------------------------------------------

#define NX 4096
#define NY 8192
#define INV_EPS 10.0f

#define CHW 32        // row-pass chunk width (columns per buffer)
#define LDSTRIDE 36   // padded LDS row stride (conflict-free WMMA-A reads)

typedef __attribute__((ext_vector_type(2))) float v2f;
typedef __attribute__((ext_vector_type(8))) float v8f;
typedef __attribute__((ext_vector_type(4))) int v4i;

#define HAVE_WMMA4 0
#define HAVE_ASYNC128 0
#define HAVE_WAIT_ASYNC 0

#if defined(__has_builtin)
#if __has_builtin(__builtin_amdgcn_wmma_f32_16x16x4_f32)
#undef HAVE_WMMA4
#define HAVE_WMMA4 1
#endif
#if __has_builtin(__builtin_amdgcn_global_load_async_to_lds_b128)
#undef HAVE_ASYNC128
#define HAVE_ASYNC128 1
#endif
#if __has_builtin(__builtin_amdgcn_s_wait_asynccnt)
#undef HAVE_WAIT_ASYNC
#define HAVE_WAIT_ASYNC 1
#endif
#endif

// Builtin signature (probe-confirmed via hipcc diagnostic): param0 = AS1
// (global) v4i*, param1 = AS3 (LDS) v4i*, then imm offset + imm cpol.
typedef __attribute__((address_space(1))) v4i GV4;
typedef __attribute__((address_space(3))) v4i LV4;

__device__ __forceinline__ v8f wmma4(v2f a, v2f b, v8f c) {
#if HAVE_WMMA4
  // (neg_a, A, neg_b, B, c_mod, C, reuse_a, reuse_b)
  return __builtin_amdgcn_wmma_f32_16x16x4_f32(false, a, false, b, (short)0, c,
                                               false, false);
#else
  c[0] += a[0] * b[0] + a[1] * b[1];  // keep-compiling fallback
  return c;
#endif
}

__device__ __forceinline__ void stage_b128(const float* gsrc, float* ldst) {
#if HAVE_ASYNC128
  __builtin_amdgcn_global_load_async_to_lds_b128((GV4*)gsrc, (LV4*)ldst, 0, 0);
#else
  *(float4*)ldst = *(const float4*)gsrc;  // ds ops are in-order within a wave
#endif
}

template <int N>
__device__ __forceinline__ void wait_async_le() {
#if HAVE_ASYNC128
#if HAVE_WAIT_ASYNC
  __builtin_amdgcn_s_wait_asynccnt(N);
#else
  asm volatile("s_wait_asynccnt %0" ::"i"(N) : "memory");
#endif
  asm volatile("" ::: "memory");
#endif
}

// Guard against WAR: make sure in-flight ds_loads from the buffer we are
// about to overwrite have completed before issuing new async writes into it.
__device__ __forceinline__ void lds_reuse_fence() {
#if HAVE_ASYNC128
  asm volatile("s_wait_dscnt 0" ::: "memory");
#endif
}

// --------------------------- E = exp(-C/eps) -------------------------------
__global__ void k_exp_neg(const float4* __restrict__ C4, float4* __restrict__ E4,
                          int n4) {
  int i = blockIdx.x * blockDim.x + threadIdx.x;
  int stride = gridDim.x * blockDim.x;
  for (; i < n4; i += stride) {
    float4 c = C4[i];
    float4 e;
    e.x = __expf(-INV_EPS * c.x);
    e.y = __expf(-INV_EPS * c.y);
    e.z = __expf(-INV_EPS * c.z);
    e.w = __expf(-INV_EPS * c.w);
    E4[i] = e;
  }
}

__global__ void k_init_wv(float* __restrict__ wv) {
  int i = blockIdx.x * blockDim.x + threadIdx.x;
  if (i < NY) wv[i] = 1.0f;
}

// ------------------------- row pass: wu = a ./ (E @ wv) --------------------
// One block per 16-row stripe; 8 waves each own a 1024-column slice.
// 16xCHW tiles double-buffered in LDS via async global->LDS DMA.
__global__ __launch_bounds__(256) void k_row(const float* __restrict__ E,
                                             const float* __restrict__ wv,
                                             float* __restrict__ wu) {
  __shared__ float tile[8][2][16 * LDSTRIDE];
  __shared__ float part[8][16];

  const int wave = threadIdx.x >> 5;
  const int lane = threadIdx.x & 31;
  const int half = lane >> 4;  // 0: K0/K1 group, 1: K2/K3 group
  const int l15 = lane & 15;
  const int r0 = blockIdx.x * 16;

  // Staging map: lane covers float4 (l&7) of row rr+(l>>3); 4 issues/lane.
  const int st_row = lane >> 3;        // 0..3
  const int st_col = (lane & 7) * 4;   // 0,4,...,28
  const float* gstripe = E + (size_t)r0 * NY;

  union {
    v8f v;
    float f[8];
  } acc;
  v8f z = {0.f, 0.f, 0.f, 0.f, 0.f, 0.f, 0.f, 0.f};
  acc.v = z;

  const int cbeg = wave * (NY / 8);
  const int kofs = half * 2;

#define STAGE(cb_, buf_)                                                     \
  do {                                                                       \
    float* dst_ = tile[wave][buf_];                                          \
    const float* src_ = gstripe + (cb_);                                     \
    for (int rr_ = 0; rr_ < 16; rr_ += 4) {                                  \
      int row_ = rr_ + st_row;                                               \
      stage_b128(src_ + (size_t)row_ * NY + st_col,                          \
                 dst_ + row_ * LDSTRIDE + st_col);                           \
    }                                                                        \
  } while (0)

  const int nchunk = (NY / 8) / CHW;  // 32
  int cur = 0;
  STAGE(cbeg, 0);
  for (int i = 0; i < nchunk; ++i) {
    const int cb = cbeg + i * CHW;
    if (i + 1 < nchunk) {
      lds_reuse_fence();
      STAGE(cb + CHW, cur ^ 1);
      wait_async_le<4>();  // current buffer's 4 transfers retired (in-order)
    } else {
      wait_async_le<0>();
    }
    // 8 WMMAs: A = 16x4 E-tile, B[k][j] = wv[cb+k] (broadcast across N).
    // D[i][j] += sum_k E[i][k]*wv[k] -> partial row-sums, replicated over j.
    const float* arow = tile[wave][cur] + l15 * LDSTRIDE + kofs;
    for (int k = 0; k < CHW; k += 4) {
      v2f a = *(const v2f*)(arow + k);
      v2f b;
      b[0] = wv[cb + k + kofs];
      b[1] = wv[cb + k + kofs + 1];
      acc.v = wmma4(a, b, acc.v);
    }
    cur ^= 1;
  }
#undef STAGE

  // Lane 0 holds rows 0..7 (col j=0), lane 16 holds rows 8..15.
  if (l15 == 0) {
    int rb = half * 8;
    for (int m = 0; m < 8; ++m) part[wave][rb + m] = acc.f[m];
  }
  __syncthreads();
  if (threadIdx.x < 16) {
    float s = 0.f;
    for (int w = 0; w < 8; ++w) s += part[w][threadIdx.x];
    const float a_eff = 1.0f / NX + 1e-8f;  // reference adds 1e-8 inside log
    wu[r0 + threadIdx.x] = a_eff / s;
  }
}

// ------------------------ col pass: wv = b ./ (E' @ wu) --------------------
// One block per 16-column stripe; 8 waves each own a 512-row slice.
// B operand loaded straight from global (2x 64B segments per load, tiled by
// adjacent blocks), A = wu broadcast.
__global__ __launch_bounds__(256) void k_col(const float* __restrict__ E,
                                             const float* __restrict__ wu,
                                             float* __restrict__ wv) {
  __shared__ float part[8][16];

  const int wave = threadIdx.x >> 5;
  const int lane = threadIdx.x & 31;
  const int half = lane >> 4;
  const int l15 = lane & 15;
  const int c0 = blockIdx.x * 16;
  const int rofs = half * 2;

  union {
    v8f v;
    float f[8];
  } acc;
  v8f z = {0.f, 0.f, 0.f, 0.f, 0.f, 0.f, 0.f, 0.f};
  acc.v = z;

  const int rbeg = wave * (NX / 8);
  for (int r = rbeg; r < rbeg + (NX / 8); r += 4) {
    const float* p = E + (size_t)(r + rofs) * NY + (c0 + l15);
    v2f b;  // B[k][j] = E[r+k][c0+j]
    b[0] = p[0];
    b[1] = p[NY];
    v2f a;  // A[i][k] = wu[r+k] (broadcast across M)
    a[0] = wu[r + rofs];
    a[1] = wu[r + rofs + 1];
    // D[i][j] += sum_k wu[r+k]*E[r+k][c0+j] -> partial col-sums, repl. over i.
    acc.v = wmma4(a, b, acc.v);
  }

  if (lane < 16) part[wave][lane] = acc.f[0];  // row M=0, col j=lane
  __syncthreads();
  if (threadIdx.x < 16) {
    float s = 0.f;
    for (int w = 0; w < 8; ++w) s += part[w][threadIdx.x];
    const float b_eff = 1.0f / NY + 1e-8f;
    wv[c0 + threadIdx.x] = b_eff / s;
  }
}

// ---------------------- pi = diag(wu) * E * diag(wv) -----------------------
__global__ void k_final(float4* __restrict__ E4, const float* __restrict__ wu,
                        const float4* __restrict__ wv4, int n4) {
  int i = blockIdx.x * blockDim.x + threadIdx.x;
  int stride = gridDim.x * blockDim.x;
  const int ny4 = NY / 4;
  for (; i < n4; i += stride) {
    int row = i / ny4;
    int j4 = i - row * ny4;
    float4 e = E4[i];
    float4 w = wv4[j4];
    float u = wu[row];
    e.x *= u * w.x;
    e.y *= u * w.y;
    e.z *= u * w.z;
    e.w *= u * w.w;
    E4[i] = e;
  }
}

extern "C" void kernel_launch(void* const* d_in, const int* in_sizes, int n_in,
                              void* d_out, int out_size, void* d_ws,
                              size_t ws_size, hipStream_t stream) {
  (void)in_sizes;
  (void)n_in;
  (void)out_size;
  (void)ws_size;
  const float* C = (const float*)d_in[0];
  float* E = (float*)d_out;  // 4096*8192 f32: holds E during iters, pi at end
  float* wu = (float*)d_ws;  // [NX]
  float* wv = wu + NX;       // [NY]

  const int n4 = NX * NY / 4;
  k_exp_neg<<<4096, 256, 0, stream>>>((const float4*)C, (float4*)E, n4);
  k_init_wv<<<NY / 256, 256, 0, stream>>>(wv);

  for (int t = 0; t < 100; ++t) {
    k_row<<<NX / 16, 256, 0, stream>>>(E, wv, wu);
    k_col<<<NY / 16, 256, 0, stream>>>(E, wu, wv);
  }

  k_final<<<4096, 256, 0, stream>>>((float4*)E, wu, (const float4*)wv, n4);
}